// LightGCN_25125558681787
// MI455X (gfx1250) — compile-verified
//
#include <hip/hip_runtime.h>
#include <hip/hip_bf16.h>
#include <stdint.h>

#define EMB        64
#define TILE       1024
#define THREADS    256
#define EPW        (TILE / (THREADS / 32))   // 128 edges per wave

typedef int v4i __attribute__((ext_vector_type(4)));
typedef __attribute__((address_space(1))) v4i* gptr_v4i;
typedef __attribute__((address_space(3))) v4i* lptr_v4i;

// ---------- CDNA5 async global->LDS copy (ASYNCcnt path), NT hint ----------
__device__ __forceinline__ void async_cp16_nt(const void* g, void* l) {
#if __has_builtin(__builtin_amdgcn_global_load_async_to_lds_b128)
  // cpol = 1 -> TH_LOAD_NT: stream edge data through without evicting the
  // L2-resident node embeddings (cur/nxt/acc ~154MB of the 192MB L2).
  __builtin_amdgcn_global_load_async_to_lds_b128(
      (gptr_v4i)g, (lptr_v4i)l, /*offset=*/0, /*cpol=*/1);
#else
  unsigned loff = (unsigned)(uintptr_t)(__attribute__((address_space(3))) void*)l;
  uint64_t ga   = (uint64_t)(uintptr_t)g;
  asm volatile("global_load_async_to_lds_b128 %0, %1, off th:TH_LOAD_NT"
               :: "v"(loff), "v"(ga) : "memory");
#endif
}

__device__ __forceinline__ void wait_async0() {
#if __has_builtin(__builtin_amdgcn_s_wait_asynccnt)
  __builtin_amdgcn_s_wait_asynccnt(0);
#else
  asm volatile("s_wait_asynccnt 0" ::: "memory");
#endif
}

// ---------- init: cur = acc = concat(user_emb, item_emb); nxt = 0 ----------
__global__ __launch_bounds__(THREADS) void lgcn_init(
    const float4* __restrict__ ue, const float4* __restrict__ ie,
    float4* __restrict__ cur, float4* __restrict__ acc,
    float4* __restrict__ nxt, int n_user4, int n4) {
  int i = blockIdx.x * blockDim.x + threadIdx.x;
  if (i >= n4) return;
  float4 v = (i < n_user4) ? ue[i] : ie[i - n_user4];
  cur[i] = v;
  acc[i] = v;
  nxt[i] = make_float4(0.f, 0.f, 0.f, 0.f);
}

// ---------- acc += x; z = 0 (z = just-consumed buffer -> next scatter target) ----------
__global__ __launch_bounds__(THREADS) void lgcn_accum_zero(
    float4* __restrict__ acc, const float4* __restrict__ x,
    float4* __restrict__ z, int n4) {
  int i = blockIdx.x * blockDim.x + threadIdx.x;
  if (i >= n4) return;
  float4 a = acc[i];
  const float4 b = x[i];
  a.x += b.x; a.y += b.y; a.z += b.z; a.w += b.w;
  acc[i] = a;
  z[i] = make_float4(0.f, 0.f, 0.f, 0.f);
}

__global__ __launch_bounds__(THREADS) void lgcn_finish(
    float4* __restrict__ acc, const float4* __restrict__ x, int n4, float s) {
  int i = blockIdx.x * blockDim.x + threadIdx.x;
  if (i >= n4) return;
  float4 a = acc[i];
  const float4 b = x[i];
  a.x = (a.x + b.x) * s; a.y = (a.y + b.y) * s;
  a.z = (a.z + b.z) * s; a.w = (a.w + b.w) * s;
  acc[i] = a;
}

// ---------- scatter: next[row] += val * cur[col], one edge per wave32 ----------
__global__ __launch_bounds__(THREADS) void lgcn_scatter(
    const int* __restrict__ erow, const int* __restrict__ ecol,
    const float* __restrict__ eval_, const float* __restrict__ cur,
    float* __restrict__ nxt, int n_edges) {
  __shared__ int   s_row[TILE];
  __shared__ int   s_col[TILE];
  __shared__ float s_val[TILE];
  const int t    = threadIdx.x;
  const int wave = t >> 5;
  const int lane = t & 31;
  const int n_tiles = (n_edges + TILE - 1) / TILE;

  const float* curL = cur + (lane << 1);   // per-lane column base
  float*       nxtL = nxt + (lane << 1);   // per-lane row base

  for (int tile = blockIdx.x; tile < n_tiles; tile += gridDim.x) {
    const int base = tile * TILE;
    int cnt = n_edges - base;
    if (cnt > TILE) cnt = TILE;

    if (cnt == TILE) {
      // stage 1024 edges (12 KB) into LDS with async B128 copies: 4 ints/lane/array
      const int o = t * 4;
      async_cp16_nt(erow  + base + o, s_row + o);
      async_cp16_nt(ecol  + base + o, s_col + o);
      async_cp16_nt(eval_ + base + o, s_val + o);
      wait_async0();
    } else {
      for (int k = t; k < cnt; k += THREADS) {
        s_row[k] = erow[base + k];
        s_col[k] = ecol[base + k];
        s_val[k] = eval_[base + k];
      }
    }
    __syncthreads();

    const int jBeg = wave * EPW;
    int jEnd = jBeg + EPW;
    if (jEnd > cnt) jEnd = cnt;
#pragma unroll 4
    for (int j = jBeg; j < jEnd; ++j) {
      const int   r = s_row[j];
      const int   c = s_col[j];
      const float v = s_val[j];
      // 32 lanes x float2 = the full 64-float embedding, 256B contiguous
      const float2 x = *(const float2*)(curL + (c << 6));
      // Single-instruction no-return FP32 atomics (STOREcnt path), fused pair
      // via offset:4, device scope (L2 = coherence point). No "memory"
      // clobber: lets the compiler hoist later gathers above these atomics.
      asm volatile(
          "global_atomic_add_f32 %0, %1, off scope:SCOPE_DEV\n\t"
          "global_atomic_add_f32 %0, %2, off offset:4 scope:SCOPE_DEV"
          :: "v"((uint64_t)(uintptr_t)(nxtL + (r << 6))),
             "v"(v * x.x), "v"(v * x.y));
    }
    __syncthreads();
  }
}

extern "C" void kernel_launch(void* const* d_in, const int* in_sizes, int n_in,
                              void* d_out, int out_size, void* d_ws, size_t ws_size,
                              hipStream_t stream) {
  const float* ue = (const float*)d_in[0];   // user_emb  [100000*64]
  const float* ie = (const float*)d_in[1];   // item_emb  [100000*64]
  const int*   er = (const int*)d_in[2];     // edge_row  [6400000]
  const int*   ec = (const int*)d_in[3];     // edge_col  [6400000]
  const float* ev = (const float*)d_in[4];   // edge_val  [6400000]

  const int nE   = in_sizes[2];
  const int nU4  = in_sizes[0] / 4;
  const int n4   = out_size / 4;             // (N_USERS+N_ITEMS)*64/4 float4s
  const size_t nodeElems = (size_t)out_size; // N*64 floats

  float* cur = (float*)d_ws;                 // 51.2 MB
  float* nxt = cur + nodeElems;              // 51.2 MB
  float* acc = (float*)d_out;

  dim3 blk(THREADS);
  dim3 gE((unsigned)((n4 + THREADS - 1) / THREADS));
  const int tiles = (nE + TILE - 1) / TILE;
  dim3 gS((unsigned)(tiles < 65535 ? tiles : 65535));

  // cur = acc = emb, nxt = 0
  lgcn_init<<<gE, blk, 0, stream>>>((const float4*)ue, (const float4*)ie,
                                    (float4*)cur, (float4*)acc, (float4*)nxt,
                                    nU4, n4);
  for (int l = 0; l < 3; ++l) {
    lgcn_scatter<<<gS, blk, 0, stream>>>(er, ec, ev, cur, nxt, nE);
    if (l == 2) {
      lgcn_finish<<<gE, blk, 0, stream>>>((float4*)acc, (const float4*)nxt, n4, 0.25f);
    } else {
      // acc += nxt; zero old cur (it becomes the next scatter target)
      lgcn_accum_zero<<<gE, blk, 0, stream>>>((float4*)acc, (const float4*)nxt,
                                              (float4*)cur, n4);
    }
    float* tmp = cur; cur = nxt; nxt = tmp;
  }
}